// DepthMask2PointCloud_57243324121543
// MI455X (gfx1250) — compile-verified
//
#include <hip/hip_runtime.h>
#include <cmath>

#ifndef M_PI
#define M_PI 3.14159265358979323846
#endif

#define HDIM 150
#define WDIM 200
#define HW (HDIM * WDIM)
#define MAXP 5
#define MAXPTS 1024
#define SLOTS (MAXPTS + 1)

#define NBINS 2048
#define CAP 256
#define CHUNK 4096   // floats per staged chunk (16 KB)
#define NWAVES 8
#define WRANGE (HW / NWAVES)  // 3750, exact

// ---- CDNA5 feature detection (host pass sees these as absent; device pass enables) ----
#if defined(__has_builtin)
#if __has_builtin(__builtin_amdgcn_global_load_async_to_lds_b128)
#define HAVE_ASYNC 1
#endif
#if __has_builtin(__builtin_amdgcn_s_wait_asynccnt)
#define HAVE_WAITASYNC 1
#endif
#if __has_builtin(__builtin_amdgcn_ballot_w32)
#define HAVE_BALLOT32 1
#endif
#endif

typedef int v4i __attribute__((ext_vector_type(4)));

template <int N>
__device__ __forceinline__ void wait_asynccnt() {
#if defined(HAVE_WAITASYNC)
    __builtin_amdgcn_s_wait_asynccnt(N);
#elif defined(HAVE_ASYNC)
    asm volatile("s_wait_asynccnt %0" ::"i"(N) : "memory");
#endif
}

__device__ __forceinline__ void async_load16(const float* gsrc, float* ldst) {
#if defined(HAVE_ASYNC)
    // per-lane 16B async copy: global -> LDS, tracked by ASYNCcnt
    __builtin_amdgcn_global_load_async_to_lds_b128((v4i*)gsrc, (v4i*)ldst, 0, 0);
#else
    *(float4*)ldst = *(const float4*)gsrc;  // synchronous fallback
#endif
}

__device__ __forceinline__ unsigned ballot32(bool m) {
#if defined(HAVE_BALLOT32)
    return (unsigned)__builtin_amdgcn_ballot_w32(m);
#else
    return (unsigned)__ballot(m);
#endif
}

__device__ __forceinline__ int bin_of(float v) {
    int bin = (int)((v - 3.0f) * ((float)NBINS / 5.0f));
    return bin < 0 ? 0 : (bin > NBINS - 1 ? NBINS - 1 : bin);
}

__device__ __forceinline__ void issue_chunk(const float* __restrict__ dptr, int ch,
                                            float* dst, int tid) {
    const int base = ch * CHUNK;
#pragma unroll
    for (int seg = 0; seg < CHUNK / (256 * 4); ++seg) {  // 4 x (256 lanes x 16B)
        const int foff = seg * (256 * 4) + tid * 4;
        int gidx = base + foff;
        if (gidx > HW - 4) gidx = HW - 4;  // clamp ragged tail (never read past HW)
        async_load16(dptr + gidx, dst + foff);
    }
}

// ---------------- Kernel 1: per-batch IQR bounds via histogram + exact rank select ----
__global__ void __launch_bounds__(256) bounds_kernel(const float* __restrict__ x,
                                                     float* __restrict__ bounds) {
    const int b = blockIdx.x;
    const int tid = threadIdx.x;
    const float* dptr = x + (size_t)b * 3 * HW;  // channel 0: depth

    __shared__ __align__(16) float s_buf[2][CHUNK];
    __shared__ unsigned s_hist[NBINS];
    __shared__ float s_cand[4][CAP];
    __shared__ int s_cnt[4];
    __shared__ int s_bin[4];
    __shared__ int s_off[4];
    __shared__ float s_res[4];
    __shared__ float s_frac[2];
    __shared__ int s_n;

    for (int i = tid; i < NBINS; i += 256) s_hist[i] = 0u;
    if (tid < 4) { s_cnt[tid] = 0; s_res[tid] = 0.0f; }
    __syncthreads();

    // -------- Pass 1: histogram of valid depths, async-staged through LDS --------
    const int nch = (HW + CHUNK - 1) / CHUNK;
    issue_chunk(dptr, 0, &s_buf[0][0], tid);
    for (int ch = 0; ch < nch; ++ch) {
        if (ch + 1 < nch) {
            issue_chunk(dptr, ch + 1, &s_buf[(ch + 1) & 1][0], tid);
            wait_asynccnt<4>();  // this wave's previous chunk done (in-order ASYNCcnt)
        } else {
            wait_asynccnt<0>();
        }
        __syncthreads();
        const float* buf = &s_buf[ch & 1][0];
        const int base = ch * CHUNK;
        for (int i = tid; i < CHUNK; i += 256) {
            const int gi = base + i;
            if (gi < HW) {
                const float v = buf[i];
                if (v > 3.0f) atomicAdd(&s_hist[bin_of(v)], 1u);
            }
        }
        __syncthreads();
    }

    // -------- Rank -> bin localization (scalar scan, trivial) --------
    if (tid == 0) {
        int n = 0;
        for (int i = 0; i < NBINS; ++i) n += (int)s_hist[i];
        s_n = n;
        int ranks[4];
        if (n > 0) {
            const float p1 = 0.25f * (float)(n - 1);
            const float p3 = 0.75f * (float)(n - 1);
            ranks[0] = (int)floorf(p1); ranks[1] = (int)ceilf(p1);
            ranks[2] = (int)floorf(p3); ranks[3] = (int)ceilf(p3);
            s_frac[0] = p1 - floorf(p1); s_frac[1] = p3 - floorf(p3);
        } else {
            ranks[0] = ranks[1] = ranks[2] = ranks[3] = 0;
            s_frac[0] = s_frac[1] = 0.0f;
        }
        int cum = 0;
        int bptr[4] = {NBINS - 1, NBINS - 1, NBINS - 1, NBINS - 1};
        int boff[4] = {0, 0, 0, 0};
        for (int i = 0; i < NBINS; ++i) {
            const int c = (int)s_hist[i];
            for (int j = 0; j < 4; ++j)
                if (ranks[j] >= cum && ranks[j] < cum + c) { bptr[j] = i; boff[j] = ranks[j] - cum; }
            cum += c;
        }
        for (int j = 0; j < 4; ++j) { s_bin[j] = bptr[j]; s_off[j] = boff[j]; }
    }
    __syncthreads();

    const int n = s_n;
    if (n > 0) {
        // -------- Pass 2: gather candidates in the 4 target bins (L2-hot) --------
        for (int i = tid; i < HW; i += 256) {
            if (i + 4096 < HW) __builtin_prefetch(dptr + i + 4096, 0, 0);
            const float v = dptr[i];
            if (v > 3.0f) {
                const int bin = bin_of(v);
#pragma unroll
                for (int j = 0; j < 4; ++j) {
                    if (bin == s_bin[j]) {
                        const int pos = atomicAdd(&s_cnt[j], 1);
                        if (pos < CAP) s_cand[j][pos] = v;
                    }
                }
            }
        }
        __syncthreads();
        // exact k-th smallest within each bin via rank counting (m is tiny)
        for (int j = 0; j < 4; ++j) {
            int m = s_cnt[j]; if (m > CAP) m = CAP;
            int target = s_off[j]; if (target >= m) target = m - 1;
            for (int i = tid; i < m; i += 256) {
                const float vi = s_cand[j][i];
                int r = 0;
                for (int k = 0; k < m; ++k) {
                    const float vk = s_cand[j][k];
                    r += (vk < vi) || (vk == vi && k < i);
                }
                if (r == target) s_res[j] = vi;  // unique writer (r is a permutation)
            }
        }
        __syncthreads();
    }

    if (tid == 0) {
        float lower, upper;
        if (n > 0) {
            const float q1 = s_res[0] + s_frac[0] * (s_res[1] - s_res[0]);
            const float q3 = s_res[2] + s_frac[1] * (s_res[3] - s_res[2]);
            const float iqr = q3 - q1;
            lower = q1 - 1.5f * iqr;
            upper = q3 + 1.5f * iqr;
        } else {
            lower = INFINITY; upper = -INFINITY;
        }
        bounds[2 * b] = lower;
        bounds[2 * b + 1] = upper;
    }
}

// ------- Kernel 2: per (batch, person); 8-wave parallel ordered ballot compaction ----
__global__ void __launch_bounds__(256) select_kernel(const float* __restrict__ x,
                                                     const float* __restrict__ bounds,
                                                     float* __restrict__ out,
                                                     float inv_fx, float inv_fy) {
    const int b = blockIdx.x / MAXP;
    const int p = blockIdx.x % MAXP;
    const float person = (float)(p + 1);
    const int tid = threadIdx.x;
    const int wave = tid >> 5;
    const int lane = tid & 31;
    const float* dptr = x + (size_t)b * 3 * HW;
    const float* iptr = dptr + HW;
    const float lower = bounds[2 * b];
    const float upper = bounds[2 * b + 1];
    float* outb = out + (size_t)b * (3 * MAXP * SLOTS) + (size_t)p * SLOTS;

    // zero our output slab (d_out is poisoned)
    for (int i = tid; i < 3 * SLOTS; i += 256) {
        const int c = i / SLOTS, s = i % SLOTS;
        outb[(size_t)c * (MAXP * SLOTS) + s] = 0.0f;
    }

    __shared__ int s_wcnt[NWAVES];
    __shared__ int s_woff[NWAVES];
    __shared__ int s_n;

    const int rbeg = wave * WRANGE;
    const int rend = rbeg + WRANGE;
    const unsigned lmask = (lane == 0) ? 0u : (0xFFFFFFFFu >> (32 - lane));

    // -------- Pass A: per-wave count over contiguous range --------
    int c = 0;
    for (int base = rbeg; base < rend; base += 32) {
        const int i = base + lane;
        bool m = false;
        if (i < rend) {
            if (i + 1024 < rend) __builtin_prefetch(dptr + i + 1024, 0, 0);
            const float dv = dptr[i];
            const float iv = rintf(iptr[i]);
            m = (dv > 3.0f) && (iv == person) && (dv > lower) && (dv < upper);
        }
        c += __builtin_popcount(ballot32(m));
    }
    if (lane == 0) s_wcnt[wave] = c;
    __syncthreads();
    if (tid == 0) {
        int acc = 0;
        for (int w = 0; w < NWAVES; ++w) { s_woff[w] = acc; acc += s_wcnt[w]; }
        s_n = acc;
    }
    __syncthreads();
    const int n_pts = s_n;
    if (tid == 0) outb[MAXPTS] = (n_pts > 0) ? 1.0f : 0.0f;  // channel-0 flag slot

    // -------- Pass B: all 8 waves emit in parallel, globally index-ordered --------
    int o0 = s_woff[wave];
    for (int base = rbeg; base < rend; base += 32) {
        const int i = base + lane;
        float dv = 0.0f;
        bool m = false;
        if (i < rend) {
            dv = dptr[i];
            const float iv = rintf(iptr[i]);
            m = (dv > 3.0f) && (iv == person) && (dv > lower) && (dv < upper);
        }
        const unsigned bal = ballot32(m);
        if (m) {
            const int o = o0 + __builtin_popcount(bal & lmask);
            bool keep; int slot;
            if (n_pts <= MAXPTS) {
                keep = true; slot = o;
            } else {
                // exactly-1024 deterministic strided subsample
                const unsigned s0 = ((unsigned)o * (unsigned)MAXPTS) / (unsigned)n_pts;
                const unsigned sp = (o == 0) ? 0xFFFFFFFFu
                                             : (((unsigned)(o - 1) * (unsigned)MAXPTS) / (unsigned)n_pts);
                keep = (s0 != sp); slot = (int)s0;
            }
            if (keep) {
                const int hh = i / WDIM, ww = i - hh * WDIM;
                const float xc = ((float)ww - (float)WDIM * 0.5f) * inv_fx;
                const float yc = ((float)hh - (float)HDIM * 0.5f) * inv_fy;
                outb[slot] = xc * dv;
                outb[MAXP * SLOTS + slot] = yc * dv;
                outb[2 * MAXP * SLOTS + slot] = dv;
            }
        }
        o0 += __builtin_popcount(bal);
    }
}

extern "C" void kernel_launch(void* const* d_in, const int* in_sizes, int n_in,
                              void* d_out, int out_size, void* d_ws, size_t ws_size,
                              hipStream_t stream) {
    const float* x = (const float*)d_in[0];
    float* out = (float*)d_out;
    float* bounds = (float*)d_ws;  // B*2 floats
    const int b = in_sizes[0] / (3 * HW);

    const double fx = (double)WDIM / (2.0 * tan(81.0 * M_PI / 180.0 / 2.0));
    const double fy = (double)HDIM / (2.0 * tan(59.0 * M_PI / 180.0 / 2.0));
    const float inv_fx = (float)(1.0 / fx);
    const float inv_fy = (float)(1.0 / fy);

    bounds_kernel<<<b, 256, 0, stream>>>(x, bounds);
    select_kernel<<<b * MAXP, 256, 0, stream>>>(x, bounds, out, inv_fx, inv_fy);
}